// MultiHeadAttention_7782480741069
// MI455X (gfx1250) — compile-verified
//
#include <hip/hip_runtime.h>
#include <hip/hip_bf16.h>
#include <math.h>

// MI455X / gfx1250, wave32. Full-precision matrix path: V_WMMA_F32_16X16X4_F32.
typedef __attribute__((ext_vector_type(2))) float v2f;
typedef __attribute__((ext_vector_type(8))) float v8f;

#define B_   4
#define S_   1024
#define D_   512
#define H_   8
#define DH_  64
#define BH_  (B_ * H_)                 // 32
#define NROW (BH_ * S_)                // 32768 score rows
#define NTOT ((size_t)NROW * S_)       // 33,554,432 score elements
#define NBUCKET (1u << 22)             // 22-bit key buckets (16 MB histogram)
#define NCHUNK  (NBUCKET / 1024)       // 4096

// ---------------------------------------------------------------- K1: norms
// One wave32 per (b,h,s): each lane sums 2 of the 64 head elements, then
// a 5-step shuffle-xor reduction. lenq/lenk indexed [bh*S + s].
__global__ void norms_kernel(const float* __restrict__ q, const float* __restrict__ k,
                             float* __restrict__ lenq, float* __restrict__ lenk) {
    int wave = (blockIdx.x * blockDim.x + threadIdx.x) >> 5;
    int lane = threadIdx.x & 31;
    int bh = wave >> 10;          // b*H + h
    int s  = wave & 1023;
    int b  = bh >> 3, h = bh & 7;
    const float* qp = q + ((size_t)(b * S_ + s)) * D_ + h * DH_ + lane * 2;
    const float* kp = k + ((size_t)(b * S_ + s)) * D_ + h * DH_ + lane * 2;
    float2 qv = *(const float2*)qp;
    float2 kv = *(const float2*)kp;
    float sq = qv.x * qv.x + qv.y * qv.y;
    float sk = kv.x * kv.x + kv.y * kv.y;
    #pragma unroll
    for (int m = 16; m >= 1; m >>= 1) {
        sq += __shfl_xor(sq, m, 32);
        sk += __shfl_xor(sk, m, 32);
    }
    if (lane == 0) {
        lenq[wave] = sqrtf(sq) + 1e-5f;
        lenk[wave] = sqrtf(sk) + 1e-5f;
    }
}

// ------------------------------------------------------------ K0: zero hist
__global__ void zero_hist_kernel(unsigned* __restrict__ hist) {
    unsigned i = blockIdx.x * blockDim.x + threadIdx.x;
    if (i < NBUCKET) hist[i] = 0u;
}

// ------------------------------------------- K2: scores via f32 WMMA + hist
// One wave per 16x16 output tile of score[bh] (1024x1024). K-loop: 16 steps
// of V_WMMA_F32_16X16X4_F32. A fragment (16x4 f32): lane<16 holds M=lane,
// K={k,k+1}; lane>=16 holds M=lane-16, K={k+2,k+3}  -> contiguous float2.
// B fragment (4x16): lane<16 holds N=lane, K-rows {k,k+1}; lanes 16-31 rows
// {k+2,k+3}. B[kk,n] = key[t0+n, kk] (score = Q K^T over head dim).
// Epilogue: cosine-normalize, store f32 score, bucket-histogram |score|.
__global__ void score_wmma_kernel(const float* __restrict__ q, const float* __restrict__ k,
                                  const float* __restrict__ lenq, const float* __restrict__ lenk,
                                  float* __restrict__ score, unsigned* __restrict__ hist) {
    int wave = (blockIdx.x * blockDim.x + threadIdx.x) >> 5;
    int lane = threadIdx.x & 31;
    int bh   = wave >> 12;          // 32 heads
    int srow = (wave >> 6) & 63;    // 64 M-tiles
    int tcol = wave & 63;           // 64 N-tiles (consecutive waves share A rows)
    int b = bh >> 3, h = bh & 7;
    int s0 = srow * 16, t0 = tcol * 16;
    int half = lane >> 4;           // 0: K+0/1, 1: K+2/3
    int lml  = lane & 15;
    const float* qbase = q + ((size_t)(b * S_ + s0 + lml)) * D_ + h * DH_;
    const float* kbase = k + ((size_t)(b * S_ + t0 + lml)) * D_ + h * DH_;

    v8f acc = {};
    #pragma unroll
    for (int k0 = 0; k0 < DH_; k0 += 4) {
        int kk = k0 + half * 2;
        float2 av = *(const float2*)(qbase + kk);   // A: M=lml, K=kk..kk+1
        float2 bv = *(const float2*)(kbase + kk);   // B: N=lml, K=kk..kk+1
        v2f a;  a.x  = av.x; a.y  = av.y;
        v2f bb; bb.x = bv.x; bb.y = bv.y;
        acc = __builtin_amdgcn_wmma_f32_16x16x4_f32(false, a, false, bb,
                                                    (short)0, acc, false, false);
    }
    // D layout: VGPR r -> (M = r + half*8, N = lml)
    float lk = lenk[bh * S_ + t0 + lml];
    #pragma unroll
    for (int r = 0; r < 8; r++) {
        int m = r + half * 8;
        float lq = lenq[bh * S_ + s0 + m];
        float v  = acc[r] / (lq * lk);
        size_t idx = ((size_t)(bh * S_ + s0 + m)) * S_ + (size_t)(t0 + lml);
        score[idx] = v;
        unsigned key    = __float_as_uint(fabsf(v));   // monotonic for x>=0
        unsigned bucket = key >> 9;                    // top 22 of 31 bits
        atomicAdd(&hist[bucket], 1u);
    }
}

// ------------------------------------------------- K3a: per-chunk histogram sums
__global__ void chunk_sum_kernel(const unsigned* __restrict__ hist,
                                 unsigned* __restrict__ chunkSum) {
    __shared__ unsigned sm[256];
    int c = blockIdx.x, t = threadIdx.x;
    const unsigned* hp = hist + (size_t)c * 1024 + t * 4;
    sm[t] = hp[0] + hp[1] + hp[2] + hp[3];
    __syncthreads();
    for (int o = 128; o >= 1; o >>= 1) {
        if (t < o) sm[t] += sm[t + o];
        __syncthreads();
    }
    if (t == 0) chunkSum[c] = sm[0];
}

// -------------------------------- K3b: exclusive suffix over 4096 chunk sums
__global__ void chunk_suffix_kernel(const unsigned* __restrict__ chunkSum,
                                    unsigned* __restrict__ chunkSuf) {
    unsigned run = 0;
    for (int c = NCHUNK - 1; c >= 0; c--) { chunkSuf[c] = run; run += chunkSum[c]; }
}

// ------------- K3c: base[b] = #elements in strictly-greater buckets; reset hist
// In-block exclusive suffix scan over the 1024 buckets of a chunk
// (Hillis-Steele on the reversed array), plus the chunk suffix offset.
__global__ void bucket_base_kernel(unsigned* __restrict__ hist,
                                   const unsigned* __restrict__ chunkSuf,
                                   unsigned* __restrict__ base) {
    __shared__ unsigned buf[2][1024];
    int c = blockIdx.x, t = threadIdx.x;
    size_t b = (size_t)c * 1024 + t;
    unsigned v = hist[b];
    int r = 1023 - t;              // reversed position
    buf[0][r] = v;
    __syncthreads();
    int cur = 0;
    for (int off = 1; off < 1024; off <<= 1) {
        unsigned x = buf[cur][r];
        if (r >= off) x += buf[cur][r - off];
        buf[cur ^ 1][r] = x;
        __syncthreads();
        cur ^= 1;
    }
    unsigned sufIncl = buf[cur][r];            // sum over buckets >= b in chunk
    base[b] = chunkSuf[c] + (sufIncl - v);     // strictly greater
    hist[b] = 0u;                              // reused as rank cursor in K4
}

// ---------------------- K4: assign global descending rank, sc = -log((r+1)/N)*sign
__global__ void rank_kernel(float* __restrict__ score,
                            const unsigned* __restrict__ base,
                            unsigned* __restrict__ cursor) {
    size_t i = (size_t)blockIdx.x * blockDim.x + threadIdx.x;
    float s = score[i];
    unsigned key    = __float_as_uint(fabsf(s));
    unsigned bucket = key >> 9;
    unsigned r = base[bucket] + atomicAdd(&cursor[bucket], 1u);
    float prob = ((float)r + 1.0f) * (1.0f / (float)NTOT);
    float val  = -logf(prob);                  // >= 0
    score[i] = (s > 0.0f) ? val : (s < 0.0f ? -val : 0.0f);
}

// ----------------------------------------------- K5: per-row L1 sums of sc
__global__ void rowsum_kernel(const float* __restrict__ sc, float* __restrict__ rowSum) {
    __shared__ float sm[256];
    int row = blockIdx.x, t = threadIdx.x;
    const float* p = sc + (size_t)row * S_ + t * 4;
    sm[t] = fabsf(p[0]) + fabsf(p[1]) + fabsf(p[2]) + fabsf(p[3]);
    __syncthreads();
    for (int o = 128; o >= 1; o >>= 1) {
        if (t < o) sm[t] += sm[t + o];
        __syncthreads();
    }
    if (t == 0) rowSum[row] = sm[0];
}

// --------------------------- K6: out = (sc/rowSum) @ V via f32 WMMA, K=1024
// One wave per 16x16 tile of out[bh] (1024 x 64). B[kk,n] = V[b, t=kk, h*64+n]
// (column-strided loads, stride D). Row normalization folded into epilogue.
__global__ void out_wmma_kernel(const float* __restrict__ sc, const float* __restrict__ vv,
                                const float* __restrict__ rowSum, float* __restrict__ out) {
    int wave = (blockIdx.x * blockDim.x + threadIdx.x) >> 5;
    int lane = threadIdx.x & 31;
    int bh   = wave >> 8;           // 32 heads
    int srow = (wave >> 2) & 63;    // 64 M-tiles
    int ncol = wave & 3;            // 4 N-tiles
    int b = bh >> 3, h = bh & 7;
    int s0 = srow * 16, n0 = ncol * 16;
    int half = lane >> 4, lml = lane & 15;
    const float* abase = sc + ((size_t)(bh * S_ + s0 + lml)) * S_;
    const float* vbase = vv + ((size_t)(b * S_)) * D_ + h * DH_ + n0 + lml;

    v8f acc = {};
    for (int k0 = 0; k0 < S_; k0 += 4) {
        int kk = k0 + half * 2;
        float2 av = *(const float2*)(abase + kk);       // A: M=lml, K=kk..kk+1
        v2f a;  a.x  = av.x; a.y  = av.y;
        v2f bb;                                         // B: N=lml, rows kk..kk+1
        bb.x = vbase[(size_t)kk * D_];
        bb.y = vbase[(size_t)(kk + 1) * D_];
        acc = __builtin_amdgcn_wmma_f32_16x16x4_f32(false, a, false, bb,
                                                    (short)0, acc, false, false);
    }
    #pragma unroll
    for (int r = 0; r < 8; r++) {
        int m = r + half * 8;
        float inv = 1.0f / rowSum[bh * S_ + s0 + m];
        out[((size_t)(b * S_ + s0 + m)) * D_ + h * DH_ + n0 + lml] = acc[r] * inv;
    }
}

// ---------------------------------------------------------------- launcher
extern "C" void kernel_launch(void* const* d_in, const int* in_sizes, int n_in,
                              void* d_out, int out_size, void* d_ws, size_t ws_size,
                              hipStream_t stream) {
    const float* q = (const float*)d_in[0];
    const float* k = (const float*)d_in[1];
    const float* v = (const float*)d_in[2];
    float* out = (float*)d_out;

    // Workspace layout (~167 MB total; score buffer fits in MI455X's 192MB L2)
    char* ws = (char*)d_ws;
    size_t off = 0;
    float*    score    = (float*)(ws + off); off += NTOT * sizeof(float);        // 134 MB
    unsigned* hist     = (unsigned*)(ws + off); off += (size_t)NBUCKET * 4;      // 16 MB
    unsigned* base     = (unsigned*)(ws + off); off += (size_t)NBUCKET * 4;      // 16 MB
    unsigned* chunkSum = (unsigned*)(ws + off); off += (size_t)NCHUNK * 4;
    unsigned* chunkSuf = (unsigned*)(ws + off); off += (size_t)NCHUNK * 4;
    float*    lenq     = (float*)(ws + off); off += (size_t)NROW * 4;
    float*    lenk     = (float*)(ws + off); off += (size_t)NROW * 4;
    float*    rowSum   = (float*)(ws + off); off += (size_t)NROW * 4;

    zero_hist_kernel<<<NBUCKET / 256, 256, 0, stream>>>(hist);
    norms_kernel<<<NROW / 8, 256, 0, stream>>>(q, k, lenq, lenk);
    // 32 heads * 64 * 64 tiles = 131072 waves, 8 waves/block
    score_wmma_kernel<<<131072 / 8, 256, 0, stream>>>(q, k, lenq, lenk, score, hist);
    chunk_sum_kernel<<<NCHUNK, 256, 0, stream>>>(hist, chunkSum);
    chunk_suffix_kernel<<<1, 1, 0, stream>>>(chunkSum, chunkSuf);
    bucket_base_kernel<<<NCHUNK, 1024, 0, stream>>>(hist, chunkSuf, base);
    rank_kernel<<<(unsigned)(NTOT / 256), 256, 0, stream>>>(score, base, hist /*cursor*/);
    rowsum_kernel<<<NROW, 256, 0, stream>>>(score, rowSum);
    // 32 heads * 64 * 4 tiles = 8192 waves, 8 waves/block
    out_wmma_kernel<<<8192 / 8, 256, 0, stream>>>(score, v, rowSum, out);
}